// Aggregate_25297357374074
// MI455X (gfx1250) — compile-verified
//
#include <hip/hip_runtime.h>
#include <hip/hip_bf16.h>

typedef float v2f __attribute__((ext_vector_type(2)));
typedef float v8f __attribute__((ext_vector_type(8)));

#define CHANNELS 64
#define INF_VAL 1e15f

// ---------------------------------------------------------------------------
// Kernel 1: segment bounds from sorted batch_indices.
// seg_start[j] = first node index of segment j; seg_start[B] = N.
// ---------------------------------------------------------------------------
__global__ void seg_bounds_kernel(const int* __restrict__ batch,
                                  int* __restrict__ seg_start,
                                  int n, int b_count) {
    int i = blockIdx.x * blockDim.x + threadIdx.x;
    if (i >= n) return;
    int b = batch[i];
    if (i == 0) {
        for (int j = 0; j <= b; ++j) seg_start[j] = 0;
    } else {
        int pb = batch[i - 1];
        for (int j = pb + 1; j <= b; ++j) seg_start[j] = i;
    }
    if (i == n - 1) {
        for (int j = b + 1; j <= b_count; ++j) seg_start[j] = n;
    }
}

// ---------------------------------------------------------------------------
// Kernel 2: per-segment raw moments via V_WMMA_F32_16X16X4_F32.
//   A = ones(16x4)  =>  D = A*B + C accumulates column sums of B into every
//   row of D. B tile = 4 rows x 16 channels of f32 (or x^2 / x^3 / x^4).
// One block (4 waves) per segment; wave w owns channel tile [16w, 16w+16).
// B operand striping (32-bit): lanes 0-15 -> K=0 (v0), K=1 (v1);
//                              lanes 16-31 -> K=2 (v0), K=3 (v1).
// Main loop: 8 rows/iter, constant-stride pointer + immediate offsets so the
// only per-iteration address math is one 64-bit add. Masked tail keeps EXEC
// all-ones (WMMA requirement).
// ---------------------------------------------------------------------------
__global__ __launch_bounds__(128) void moments_kernel(
    const float* __restrict__ graph,
    const int* __restrict__ seg_start,
    float* __restrict__ out) {
    const int seg   = blockIdx.x;
    const int s     = seg_start[seg];
    const int e     = seg_start[seg + 1];
    const int wave  = threadIdx.x >> 5;     // 0..3 -> channel tile
    const int lane  = threadIdx.x & 31;
    const int n     = lane & 15;            // channel within tile
    const int khalf = lane >> 4;            // 0: rows +0,+1 ; 1: rows +2,+3
    const int ch    = (wave << 4) + n;      // absolute channel 0..63

    const v2f a_ones = {1.0f, 1.0f};        // A matrix = all ones
    v8f c1 = {}, c2 = {}, c3 = {}, c4 = {}; // sum x, x^2, x^3, x^4

    int r = s;
    // Per-lane base pointer: row (s + 2*khalf), channel ch.
    const float* p = graph + (size_t)(s + (khalf << 1)) * CHANNELS + ch;

    // ---- main loop: full 8-row groups, no masking, immediate offsets ----
    for (; r + 8 <= e; r += 8) {
        float x0 = p[0];                // row r + 2*khalf
        float x1 = p[CHANNELS];         // row r + 2*khalf + 1
        float y0 = p[4 * CHANNELS];     // row r + 4 + 2*khalf
        float y1 = p[5 * CHANNELS];     // row r + 5 + 2*khalf
        p += 8 * CHANNELS;

        float x0_2 = x0 * x0,     x1_2 = x1 * x1;
        float y0_2 = y0 * y0,     y1_2 = y1 * y1;
        float x0_3 = x0_2 * x0,   x1_3 = x1_2 * x1;
        float y0_3 = y0_2 * y0,   y1_3 = y1_2 * y1;
        float x0_4 = x0_2 * x0_2, x1_4 = x1_2 * x1_2;
        float y0_4 = y0_2 * y0_2, y1_4 = y1_2 * y1_2;

        v2f bx1 = {x0,   x1},   by1 = {y0,   y1};
        v2f bx2 = {x0_2, x1_2}, by2 = {y0_2, y1_2};
        v2f bx3 = {x0_3, x1_3}, by3 = {y0_3, y1_3};
        v2f bx4 = {x0_4, x1_4}, by4 = {y0_4, y1_4};

        c1 = __builtin_amdgcn_wmma_f32_16x16x4_f32(false, a_ones, false, bx1,
                                                   (short)0, c1, false, false);
        c2 = __builtin_amdgcn_wmma_f32_16x16x4_f32(false, a_ones, false, bx2,
                                                   (short)0, c2, false, false);
        c3 = __builtin_amdgcn_wmma_f32_16x16x4_f32(false, a_ones, false, bx3,
                                                   (short)0, c3, false, false);
        c4 = __builtin_amdgcn_wmma_f32_16x16x4_f32(false, a_ones, false, bx4,
                                                   (short)0, c4, false, false);
        c1 = __builtin_amdgcn_wmma_f32_16x16x4_f32(false, a_ones, false, by1,
                                                   (short)0, c1, false, false);
        c2 = __builtin_amdgcn_wmma_f32_16x16x4_f32(false, a_ones, false, by2,
                                                   (short)0, c2, false, false);
        c3 = __builtin_amdgcn_wmma_f32_16x16x4_f32(false, a_ones, false, by3,
                                                   (short)0, c3, false, false);
        c4 = __builtin_amdgcn_wmma_f32_16x16x4_f32(false, a_ones, false, by4,
                                                   (short)0, c4, false, false);
    }

    // ---- masked tail: <8 remaining rows, clamped loads keep EXEC uniform ----
    const int last = (e > 0) ? (e - 1) : 0;
    for (; r < e; r += 4) {
        int r0 = r + (khalf << 1);
        int r1 = r0 + 1;
        int rc0 = (r0 < e) ? r0 : last;
        int rc1 = (r1 < e) ? r1 : last;
        float m0 = (r0 < e) ? 1.0f : 0.0f;
        float m1 = (r1 < e) ? 1.0f : 0.0f;
        float x0 = graph[(size_t)rc0 * CHANNELS + ch] * m0;
        float x1 = graph[(size_t)rc1 * CHANNELS + ch] * m1;

        float x0_2 = x0 * x0,     x1_2 = x1 * x1;
        float x0_3 = x0_2 * x0,   x1_3 = x1_2 * x1;
        float x0_4 = x0_2 * x0_2, x1_4 = x1_2 * x1_2;

        v2f b1 = {x0,   x1};
        v2f b2 = {x0_2, x1_2};
        v2f b3 = {x0_3, x1_3};
        v2f b4 = {x0_4, x1_4};

        c1 = __builtin_amdgcn_wmma_f32_16x16x4_f32(false, a_ones, false, b1,
                                                   (short)0, c1, false, false);
        c2 = __builtin_amdgcn_wmma_f32_16x16x4_f32(false, a_ones, false, b2,
                                                   (short)0, c2, false, false);
        c3 = __builtin_amdgcn_wmma_f32_16x16x4_f32(false, a_ones, false, b3,
                                                   (short)0, c3, false, false);
        c4 = __builtin_amdgcn_wmma_f32_16x16x4_f32(false, a_ones, false, b4,
                                                   (short)0, c4, false, false);
    }

    // Every row of D holds the segment sums; row M=0 lives in VGPR0, lanes 0-15.
    if (lane < 16) {
        float cnt = (float)((e - s) > 0 ? (e - s) : 1);
        float inv = 1.0f / cnt;
        float s1 = c1[0] * inv;   // E[x]   = mean
        float s2 = c2[0] * inv;   // E[x^2]
        float s3 = c3[0] * inv;   // E[x^3]
        float s4 = c4[0] * inv;   // E[x^4]

        float mean  = s1;
        float mean2 = mean * mean;
        float m2    = s2 - mean2;                              // variance
        float m3    = s3 - 3.0f * mean * s2 + 2.0f * mean2 * mean;
        float m4    = s4 - 4.0f * mean * s3 + 6.0f * mean2 * s2
                         - 3.0f * mean2 * mean2;

        float skew = ((m3 > INF_VAL) || (m3 != m3)) ? 0.0f : m3;
        float kurt = m4 - 3.0f;
        kurt = ((kurt > INF_VAL) || (kurt != kurt)) ? -3.0f : kurt;

        float* o = out + (size_t)seg * (4 * CHANNELS);
        o[ch]                = mean;
        o[CHANNELS + ch]     = m2;
        o[2 * CHANNELS + ch] = skew;
        o[3 * CHANNELS + ch] = kurt;
    }
}

// ---------------------------------------------------------------------------
extern "C" void kernel_launch(void* const* d_in, const int* in_sizes, int n_in,
                              void* d_out, int out_size, void* d_ws, size_t ws_size,
                              hipStream_t stream) {
    const float* graph = (const float*)d_in[0];
    const int*   batch = (const int*)d_in[1];
    float*       out   = (float*)d_out;

    const int N = in_sizes[1];                 // number of nodes
    const int B = out_size / (4 * CHANNELS);   // number of segments

    int* seg_start = (int*)d_ws;               // (B+1) ints of scratch

    seg_bounds_kernel<<<(N + 255) / 256, 256, 0, stream>>>(batch, seg_start, N, B);
    moments_kernel<<<B, 128, 0, stream>>>(graph, seg_start, out);
}